// PolymorphicSNN_3281355014628
// MI455X (gfx1250) — compile-verified
//
#include <hip/hip_runtime.h>

// ---------------------------------------------------------------------------
// PolymorphicSNN forward for MI455X (gfx1250, wave32).
// Dominant cost: two fp32 GEMVs (256 MiB + 64 MiB weight streams) -> HBM-bound
// (~14us @ 23.3 TB/s). GEMVs run on the CDNA5 matrix pipe via
// V_WMMA_F32_16X16X4_F32 with the x-vector replicated across all 16 B-columns.
// Inner loop manually unrolled x4 with two independent accumulator chains so
// 8 loads are in flight per wave and adjacent WMMAs have no RAW dependency.
// ---------------------------------------------------------------------------

typedef float v2f __attribute__((ext_vector_type(2)));
typedef float v8f __attribute__((ext_vector_type(8)));

static constexpr int Nn = 8192;   // num_neurons
static constexpr int Pp = 2048;   // num_polymorphic
static constexpr int NWAVES = 8;  // waves per block (256 threads, wave32)

__device__ __forceinline__ v8f wmma_f32(v2f a, v2f b, v8f c) {
    return __builtin_amdgcn_wmma_f32_16x16x4_f32(
        /*neg_a=*/false, a, /*neg_b=*/false, b,
        /*c_mod=*/(short)0, c, /*reuse_a=*/false, /*reuse_b=*/false);
}

// y[m0..m0+15] = W[m0..m0+15, :] @ x   (K multiple of NWAVES*16)
// LEAKY=true additionally fuses the regular Leaky membrane update + spike.
template <bool LEAKY>
__global__ __launch_bounds__(NWAVES * 32)
void gemv16_wmma(const float* __restrict__ W,      // [M x K] row-major
                 const float* __restrict__ x,      // [K]
                 int K,
                 const float* __restrict__ bias,   // [M]   (LEAKY only)
                 const float* __restrict__ mem,    // [M]   (LEAKY only)
                 float* __restrict__ y_out,        // [M]   raw y, or spikes if LEAKY
                 float* __restrict__ mem_out,      // [M]   (LEAKY only)
                 float* __restrict__ comb_out,     // [M]   (LEAKY only)
                 const float* __restrict__ g_adapt)// [1]   (LEAKY only)
{
    __shared__ float red[NWAVES][16];

    const int tid  = threadIdx.x;
    const int wave = tid >> 5;
    const int lane = tid & 31;
    const int half = lane >> 4;          // ISA A-layout: VGPR0 holds K+{0,1}|K+{2,3}
    const int m0   = blockIdx.x * 16;
    const int mrow = m0 + (lane & 15);   // A-matrix: lanes 0-15 / 16-31 both map M=0..15

    const int kspan = K / NWAVES;        // 1024 for K=8192
    const int kbeg  = wave * kspan;
    const int kend  = kbeg + kspan;

    const float* wp = W + (size_t)mrow * (size_t)K + 2 * half;  // A stream
    const float* xp = x + 2 * half;                             // B stream

    v8f acc0 = {};
    v8f acc1 = {};

    for (int k = kbeg; k < kend; k += 16) {
        // Batch all 8 loads first so loadcnt waits are incremental, not a drain.
        v2f a0 = *(const v2f*)(wp + k);
        v2f a1 = *(const v2f*)(wp + k + 4);
        v2f a2 = *(const v2f*)(wp + k + 8);
        v2f a3 = *(const v2f*)(wp + k + 12);
        v2f b0 = *(const v2f*)(xp + k);
        v2f b1 = *(const v2f*)(xp + k + 4);
        v2f b2 = *(const v2f*)(xp + k + 8);
        v2f b3 = *(const v2f*)(xp + k + 12);
        // keep the weight stream ~1KB ahead (global_prefetch_b8)
        __builtin_prefetch(wp + k + 256, 0, 1);

        // Two chains: adjacent WMMAs are independent (no D->C RAW back-to-back).
        acc0 = wmma_f32(a0, b0, acc0);
        acc1 = wmma_f32(a1, b1, acc1);
        acc0 = wmma_f32(a2, b2, acc0);
        acc1 = wmma_f32(a3, b3, acc1);
    }
    v8f acc = acc0 + acc1;

    // C/D layout: VGPR v, lanes 0-15 -> (M=v, N=lane); lanes 16-31 -> (M=v+8).
    // All 16 columns are identical, so lane 0 / lane 16 each hold 8 row results.
    if (lane == 0) {
#pragma unroll
        for (int v = 0; v < 8; ++v) red[wave][v] = acc[v];
    } else if (lane == 16) {
#pragma unroll
        for (int v = 0; v < 8; ++v) red[wave][8 + v] = acc[v];
    }
    __syncthreads();

    if (tid < 16) {
        float y = 0.0f;
#pragma unroll
        for (int w = 0; w < NWAVES; ++w) y += red[w][tid];
        const int m = m0 + tid;
        if constexpr (LEAKY) {
            y += bias[m];
            const float mold  = mem[m];
            const float reset = (mold > 1.0f) ? 1.0f : 0.0f;     // detached reset
            const float mnew  = 0.5f * mold + y - reset * 1.0f;  // beta=0.5, thr=1.0
            const float spk   = ((mnew - 1.0f) > 0.0f) ? 1.0f : 0.0f;
            mem_out[m]  = mnew;
            y_out[m]    = spk;               // raw spikes -> workspace (feeds conn GEMV)
            comb_out[m] = spk * g_adapt[0];  // combined[0:N]
        } else {
            y_out[m] = y;                    // neuron_in -> workspace
        }
    }
}

// Per polymorphic neuron: Linear(1,16)->ReLU->Linear(16,3)->Softmax, 3 Leaky
// modes, probability-weighted spike mixing.
__global__ __launch_bounds__(256)
void poly_modes(const float* __restrict__ ni_ws,    // [P] neuron_in
                const float* __restrict__ pmem_in,  // [P,3]
                const float* __restrict__ W1,       // [P,16]
                const float* __restrict__ b1,       // [P,16]
                const float* __restrict__ W2,       // [P,3,16]
                const float* __restrict__ b2,       // [P,3]
                const float* __restrict__ g_adapt,  // [1]
                float* __restrict__ comb_out,       // combined + N
                float* __restrict__ pmem_out,       // [P,3]
                float* __restrict__ probs_out)      // [P,3]
{
    const float BET[3] = {0.3f, 0.7f, 0.5f};
    const float THR[3] = {1.0f, 0.8f, 1.2f};

    const int p = blockIdx.x * 256 + threadIdx.x;
    if (p >= Pp) return;

    const float ni = ni_ws[p];

    float h[16];
#pragma unroll
    for (int j = 0; j < 16; ++j) {
        const float v = fmaf(ni, W1[p * 16 + j], b1[p * 16 + j]);
        h[j] = v > 0.0f ? v : 0.0f;
    }

    float logit[3];
#pragma unroll
    for (int m = 0; m < 3; ++m) {
        float s = b2[p * 3 + m];
#pragma unroll
        for (int j = 0; j < 16; ++j)
            s = fmaf(h[j], W2[(p * 3 + m) * 16 + j], s);
        logit[m] = s;
    }

    const float mx  = fmaxf(logit[0], fmaxf(logit[1], logit[2]));
    const float e0  = __expf(logit[0] - mx);
    const float e1  = __expf(logit[1] - mx);
    const float e2  = __expf(logit[2] - mx);
    const float inv = 1.0f / (e0 + e1 + e2);
    const float pr[3] = {e0 * inv, e1 * inv, e2 * inv};

    float fin = 0.0f;
#pragma unroll
    for (int m = 0; m < 3; ++m) {
        const float po    = pmem_in[p * 3 + m];
        const float reset = (po > THR[m]) ? 1.0f : 0.0f;   // detached reset
        const float pn    = BET[m] * po + ni - reset * THR[m];
        const float spk   = ((pn - THR[m]) > 0.0f) ? 1.0f : 0.0f;
        pmem_out[p * 3 + m]  = pn;
        probs_out[p * 3 + m] = pr[m];
        fin = fmaf(spk, pr[m], fin);
    }
    comb_out[p] = fin * g_adapt[0];
}

extern "C" void kernel_launch(void* const* d_in, const int* in_sizes, int n_in,
                              void* d_out, int out_size, void* d_ws, size_t ws_size,
                              hipStream_t stream) {
    (void)in_sizes; (void)n_in; (void)out_size; (void)ws_size;

    const float* x    = (const float*)d_in[0];   // [N]
    const float* mem  = (const float*)d_in[1];   // [N]
    const float* pmem = (const float*)d_in[2];   // [P,3]
    const float* W_in = (const float*)d_in[3];   // [N,N]
    const float* b_in = (const float*)d_in[4];   // [N]
    const float* conn = (const float*)d_in[5];   // [P,N]
    const float* W1   = (const float*)d_in[6];   // [P,16]
    const float* b1   = (const float*)d_in[7];   // [P,16]
    const float* W2   = (const float*)d_in[8];   // [P,3,16]
    const float* b2   = (const float*)d_in[9];   // [P,3]
    const float* g    = (const float*)d_in[10];  // [1]

    float* out      = (float*)d_out;
    float* comb     = out;                  // [N+P]
    float* regmem   = out + (Nn + Pp);      // [N]
    float* pmem_out = regmem + Nn;          // [P,3]
    float* probs    = pmem_out + 3 * Pp;    // [P,3]

    float* spk_ws = (float*)d_ws;           // [N] regular spikes
    float* ni_ws  = spk_ws + Nn;            // [P] neuron_in

    // 1) x_proc = W_in @ x + b_in, fused Leaky + spike (HBM-bound: 256 MiB)
    gemv16_wmma<true><<<Nn / 16, NWAVES * 32, 0, stream>>>(
        W_in, x, Nn, b_in, mem, spk_ws, regmem, comb, g);

    // 2) neuron_in = conn @ reg_spk (64 MiB)
    gemv16_wmma<false><<<Pp / 16, NWAVES * 32, 0, stream>>>(
        conn, spk_ws, Nn, nullptr, nullptr, ni_ws, nullptr, nullptr, nullptr);

    // 3) per-poly-neuron MLP + softmax + 3-mode Leaky + mixing
    poly_modes<<<(Pp + 255) / 256, 256, 0, stream>>>(
        ni_ws, pmem, W1, b1, W2, b2, g, comb + Nn, pmem_out, probs);
}